// GeneratorLoss_36318243455188
// MI455X (gfx1250) — compile-verified
//
#include <hip/hip_runtime.h>
#include <hip/hip_bf16.h>

// GeneratorLoss: out = sum over (b,s) of log(clamp(pred[b,s,x[b,s]],1e-20,1)) * rewards[b,s]
// B=4, S=2048, V=50257  ->  N = 8192 gathered elements (~100 KB traffic, latency-bound).

typedef __attribute__((ext_vector_type(2))) float v2f;
typedef __attribute__((ext_vector_type(8))) float v8f;

#define VOCAB   50257
#define NPOS    8192      // B*S
#define BLOCKS  64
#define THREADS 128       // 4 wave32 per block; BLOCKS*THREADS == NPOS

// Stage 1: gather target log-prob, weight by reward, reduce per block -> partials[64]
__global__ __launch_bounds__(THREADS)
void gl_partial_kernel(const float* __restrict__ pred,
                       const int*   __restrict__ x,
                       const float* __restrict__ rewards,
                       float*       __restrict__ partials) {
    const int i = blockIdx.x * THREADS + threadIdx.x;   // 0..NPOS-1, exact fit
    const int idx = x[i];
    const unsigned long long off =
        (unsigned long long)i * (unsigned long long)VOCAB + (unsigned long long)(unsigned)idx;

    float p = pred[off];
    p = fminf(fmaxf(p, 1e-20f), 1.0f);
    float v = logf(p) * rewards[i];

    // wave32 butterfly reduction
    #pragma unroll
    for (int m = 16; m >= 1; m >>= 1)
        v += __shfl_xor(v, m, 32);

    __shared__ float sdata[THREADS / 32];
    const int wid  = threadIdx.x >> 5;
    const int lane = threadIdx.x & 31;
    if (lane == 0) sdata[wid] = v;
    __syncthreads();

    if (wid == 0) {
        float w = (lane < (THREADS / 32)) ? sdata[lane] : 0.0f;
        #pragma unroll
        for (int m = (THREADS / 32) / 2; m >= 1; m >>= 1)
            w += __shfl_xor(w, m, 32);
        if (lane == 0) partials[blockIdx.x] = w;
    }
}

// Stage 2: one wave32. Reduce 64 partials -> scalar via V_WMMA_F32_16X16X4_F32.
// A = ones(16x4), B = partials as 4x16 (2 VGPRs x 32 lanes = all 64 elements,
// in whatever lane mapping the HW uses). D[m][n] = sum_k B[k][n], so the sum
// of row M=0 (c[0] across lanes 0..15) equals the total sum of all 64 values,
// independent of the exact B operand layout. All-f32 path -> no precision loss.
__global__ __launch_bounds__(32)
void gl_final_wmma(const float* __restrict__ partials, float* __restrict__ out) {
    const int lane = threadIdx.x;   // 0..31, EXEC all ones (WMMA requirement)

    v2f b;
    b.x = partials[lane];
    b.y = partials[lane + 32];

    v2f a;
    a.x = 1.0f;
    a.y = 1.0f;

    v8f c = {};
    // 8 args: (neg_a, A, neg_b, B, c_mod, C, reuse_a, reuse_b)
    c = __builtin_amdgcn_wmma_f32_16x16x4_f32(
        /*neg_a=*/false, a, /*neg_b=*/false, b,
        /*c_mod=*/(short)0, c, /*reuse_a=*/false, /*reuse_b=*/false);

    // c[0]: lanes 0..15 hold D[M=0][N=0..15]; lanes 16..31 hold D[M=8][*].
    // Reduce within the low 16-lane group only (each row already totals the sum).
    float v = c[0];
    v += __shfl_xor(v, 1, 32);
    v += __shfl_xor(v, 2, 32);
    v += __shfl_xor(v, 4, 32);
    v += __shfl_xor(v, 8, 32);

    if (lane == 0) out[0] = v;   // unconditional write each call -> deterministic
}

extern "C" void kernel_launch(void* const* d_in, const int* in_sizes, int n_in,
                              void* d_out, int out_size, void* d_ws, size_t ws_size,
                              hipStream_t stream) {
    (void)in_sizes; (void)n_in; (void)out_size; (void)ws_size;

    const float* pred    = (const float*)d_in[0];  // (4,2048,50257) f32
    const int*   x       = (const int*)  d_in[1];  // (4,2048) int
    const float* rewards = (const float*)d_in[2];  // (4,2048) f32
    float*       out     = (float*)d_out;          // scalar f32
    float*       partials = (float*)d_ws;          // 64 floats of scratch

    gl_partial_kernel<<<BLOCKS, THREADS, 0, stream>>>(pred, x, rewards, partials);
    gl_final_wmma<<<1, 32, 0, stream>>>(partials, out);
}